// GNN_50483045597216
// MI455X (gfx1250) — compile-verified
//
#include <hip/hip_runtime.h>
#include <hip/hip_bf16.h>

// ---------------------------------------------------------------------------
// T-GCN style autoregressive GNN for MI455X (gfx1250, wave32).
// GEMMs use V_WMMA_F32_16X16X4_F32 at full fp32 precision (compute is a
// rounding error vs. the edge-scatter memory traffic). All GEMM dims are
// padded/compile-time so the WMMA loop has NO divergence and NO guards:
// fully unrolled K, immediate-offset global loads, EXEC all-ones.
// ---------------------------------------------------------------------------

typedef __attribute__((ext_vector_type(2))) float v2f;
typedef __attribute__((ext_vector_type(8))) float v8f;

#define NN      50000
#define EE      800000
#define TPAST   12
#define HID     128
#define TMPF    268
#define TMPP    272      // TMPF padded to a multiple of 16 for guard-free WMMA
#define TFUT    12

// ---------------- utility kernels ----------------

__global__ void zero_flat(float* __restrict__ p, long long n) {
    long long i = (long long)blockIdx.x * blockDim.x + threadIdx.x;
    if (i < n) p[i] = 0.0f;
}

__global__ void zero_strided(float* __restrict__ p, int nrows, int F, int ld) {
    long long i = (long long)blockIdx.x * blockDim.x + threadIdx.x;
    if (i >= (long long)nrows * F) return;
    int r = (int)(i / F), c = (int)(i % F);
    p[(size_t)r * ld + c] = 0.0f;
}

__global__ void copy_flat(float* __restrict__ d, const float* __restrict__ s, long long n) {
    long long i = (long long)blockIdx.x * blockDim.x + threadIdx.x;
    if (i < n) d[i] = s[i];
}

__global__ void copy_strided(float* __restrict__ d, int ldd,
                             const float* __restrict__ s, int lds,
                             int nrows, int F) {
    long long i = (long long)blockIdx.x * blockDim.x + threadIdx.x;
    if (i >= (long long)nrows * F) return;
    int r = (int)(i / F), c = (int)(i % F);
    d[(size_t)r * ldd + c] = s[(size_t)r * lds + c];
}

// ---------------- gcn_norm precompute ----------------

__global__ void deg_accum(const long long* __restrict__ ei,
                          const float* __restrict__ ew,
                          float* __restrict__ deg, int E) {
    int e = blockIdx.x * blockDim.x + threadIdx.x;
    if (e >= E) return;
    long long d = ei[(size_t)E + e];
    atomicAdd(&deg[d], ew[e]);
}

__global__ void dinv_kernel(const float* __restrict__ deg, float* __restrict__ dinv, int n) {
    int i = blockIdx.x * blockDim.x + threadIdx.x;
    if (i >= n) return;
    dinv[i] = rsqrtf(deg[i] + 1.0f);   // +1 for self-loop weight; always > 0
}

__global__ void norm_kernel(const long long* __restrict__ ei,
                            const float* __restrict__ ew,
                            const float* __restrict__ dinv,
                            float* __restrict__ norm, int E) {
    int e = blockIdx.x * blockDim.x + threadIdx.x;
    if (e >= E) return;
    long long s = ei[e], d = ei[(size_t)E + e];
    norm[e] = dinv[s] * ew[e] * dinv[d];
}

// ---------------- WMMA fp32 GEMM: C[M,NC] = A[M,K] @ W[K,NC] ----------------
// One wave per 16x16 C tile. All of K, NC, LDA, LDC compile-time; M%16==0,
// K%4==0, NC%16==0 -> no guards, no divergence, full unroll, immediate-offset
// loads. ISA layouts (CDNA5 §7.12.2):
//  A 16x4 : lane m=lane&15; lanes 0-15 hold K=k0,k0+1; lanes 16-31 K=k0+2,k0+3
//  B  4x16: mirrored (row striped across lanes within a VGPR)
//  D 16x16: VGPR j -> M = j + 8*(lane>>4), N = lane&15
template <int K, int NC, int LDA, int LDC>
__global__ void wmma_gemm_f32(const float* __restrict__ A,
                              const float* __restrict__ W,
                              float* __restrict__ C, int M) {
    const int lane   = threadIdx.x & 31;
    const int wave   = threadIdx.x >> 5;
    const int wpb    = blockDim.x >> 5;
    const int tilesN = NC / 16;
    const int tilesM = M / 16;
    int tile = blockIdx.x * wpb + wave;
    if (tile >= tilesM * tilesN) return;           // wave-uniform: EXEC stays full
    const int tm = tile / tilesN;
    const int tn = tile % tilesN;

    const int half = lane >> 4;                    // 0 or 1
    const int l15  = lane & 15;

    // 32-bit offset math (max offset < 2^25), then fixed pointers + imm offsets
    const float* __restrict__ ap = A + (unsigned)((tm * 16 + l15) * LDA + 2 * half);
    const float* __restrict__ bp = W + (unsigned)((2 * half) * NC + tn * 16 + l15);

    v8f acc = {};
#pragma unroll
    for (int k = 0; k < K / 4; ++k) {
        v2f a, b;
        a.x = ap[4 * k];               // contiguous pair -> global_load_b64
        a.y = ap[4 * k + 1];
        b.x = bp[(4 * k) * NC];        // imm-offset global_load_b32
        b.y = bp[(4 * k + 1) * NC];
        acc = __builtin_amdgcn_wmma_f32_16x16x4_f32(
            /*neg_a=*/false, a, /*neg_b=*/false, b,
            /*c_mod=*/(short)0, acc, /*reuse_a=*/false, /*reuse_b=*/false);
    }

    float* __restrict__ cp = C + (unsigned)((tm * 16 + 8 * half) * LDC + tn * 16 + l15);
#pragma unroll
    for (int j = 0; j < 8; ++j)
        cp[j * LDC] = acc[j];
}

// ---------------- edge scatter: out[dst] += G[src] * norm ----------------
// Feature-major thread order: consecutive threads walk one edge's feature
// row -> contiguous float4 gathers and contiguous atomics on the dst row.
template <int F>
__global__ void edge_scatter_vec4(const float* __restrict__ G, int ldg,
                                  float* __restrict__ out, int ldo,
                                  const long long* __restrict__ ei,
                                  const float* __restrict__ norm, int E) {
    constexpr int chunks = F / 4;
    long long tid = (long long)blockIdx.x * blockDim.x + threadIdx.x;
    if (tid >= (long long)E * chunks) return;
    int e  = (int)(tid / chunks);                 // div by constant
    int c4 = (int)(tid % chunks) << 2;
    long long s = ei[e], d = ei[(size_t)E + e];
    float w = norm[e];
    const float4 g = *(const float4*)(G + (size_t)s * ldg + c4);
    float* o = out + (size_t)d * ldo + c4;
    atomicAdd(o + 0, g.x * w);
    atomicAdd(o + 1, g.y * w);
    atomicAdd(o + 2, g.z * w);
    atomicAdd(o + 3, g.w * w);
}

__global__ void edge_scatter_scalar(const float* __restrict__ g,
                                    float* __restrict__ out,
                                    const long long* __restrict__ ei,
                                    const float* __restrict__ norm, int E) {
    int e = blockIdx.x * blockDim.x + threadIdx.x;
    if (e >= E) return;
    long long s = ei[e], d = ei[(size_t)E + e];
    atomicAdd(&out[d], g[s] * norm[e]);
}

// ---------------- self-loop + bias + leaky_relu (fused epilogue) ----------------
template <int F>
__global__ void finish_lrelu(float* __restrict__ out, int ldo,
                             const float* __restrict__ G, int ldg,
                             const float* __restrict__ dinv,
                             const float* __restrict__ bias, int nrows) {
    long long i = (long long)blockIdx.x * blockDim.x + threadIdx.x;
    if (i >= (long long)nrows * F) return;
    int n = (int)(i / F), c = (int)(i % F);
    float dv = dinv[n];
    float v = out[(size_t)n * ldo + c] + G[(size_t)n * ldg + c] * dv * dv + bias[c];
    out[(size_t)n * ldo + c] = (v > 0.0f) ? v : 0.01f * v;
}

// ---------------- 268 -> 1 GEMV ----------------
__global__ void gemv268(const float* __restrict__ A, const float* __restrict__ w,
                        float* __restrict__ y, int n) {
    int i = blockIdx.x * blockDim.x + threadIdx.x;
    if (i >= n) return;
    const float4* a4 = (const float4*)(A + (size_t)i * TMPF);
    const float4* w4 = (const float4*)w;
    float s = 0.0f;
#pragma unroll
    for (int k = 0; k < TMPF / 4; ++k) {
        float4 a = a4[k], ww = w4[k];
        s += a.x * ww.x + a.y * ww.y + a.z * ww.z + a.w * ww.w;
    }
    y[i] = s;
}

// ---------------- finalize: yp, emit output column t, autoregressive shift ----------------
__global__ void finalize_step(const float* __restrict__ ypacc,
                              const float* __restrict__ g3,
                              const float* __restrict__ dinv,
                              const float* __restrict__ b3,
                              const float* __restrict__ xold,
                              float* __restrict__ xnew,
                              float* __restrict__ out, int n, int t) {
    int i = blockIdx.x * blockDim.x + threadIdx.x;
    if (i >= n) return;
    float dv = dinv[i];
    float yp = ypacc[i] + g3[i] * dv * dv + b3[0];   // last conv: no lrelu
    out[(size_t)i * TFUT + t] = yp;
#pragma unroll
    for (int j = 0; j < TPAST - 1; ++j)
        xnew[(size_t)i * TPAST + j] = xold[(size_t)i * TPAST + j + 1];
    xnew[(size_t)i * TPAST + TPAST - 1] = yp;
}

// ---------------------------------------------------------------------------

static inline int cdiv_ll(long long a, long long b) { return (int)((a + b - 1) / b); }

extern "C" void kernel_launch(void* const* d_in, const int* in_sizes, int n_in,
                              void* d_out, int out_size, void* d_ws, size_t ws_size,
                              hipStream_t stream) {
    const float*      x  = (const float*)d_in[0];
    const long long*  ei = (const long long*)d_in[1];
    const float*      ew = (const float*)d_in[2];
    const float*      W0 = (const float*)d_in[3];
    const float*      b0 = (const float*)d_in[4];
    const float*      W1 = (const float*)d_in[5];
    const float*      b1 = (const float*)d_in[6];
    const float*      W2 = (const float*)d_in[7];
    const float*      b2 = (const float*)d_in[8];
    const float*      W3 = (const float*)d_in[9];
    const float*      b3 = (const float*)d_in[10];
    float*            out = (float*)d_out;
    (void)in_sizes; (void)n_in; (void)out_size; (void)ws_size;

    // workspace carve-up (all float, hipMalloc base is well-aligned)
    float* ws    = (float*)d_ws;
    float* dinv  = ws;  ws += NN;
    float* norm  = ws;  ws += EE;
    float* deg   = ws;  ws += NN;                 // reused as ypacc per step
    float* xb0   = ws;  ws += (size_t)NN * TPAST;
    float* xb1   = ws;  ws += (size_t)NN * TPAST;
    float* G     = ws;  ws += (size_t)NN * TMPP;  // pre-scatter GEMM output (ld up to 272)
    float* temp  = ws;  ws += (size_t)NN * TMPF;  // [x | h0 | h1]
    float* xt    = ws;  ws += (size_t)NN * TMPF;
    float* g3    = ws;  ws += NN;
    float* W2p   = ws;  ws += (size_t)TMPF * TMPP; // W2 padded to 268x272

    const int B = 256;  // 8 wave32 waves per block

    // --- W2 padded copy (pad cols 268..271 with zeros) ---
    zero_flat<<<cdiv_ll((long long)TMPF * TMPP, B), B, 0, stream>>>(W2p, (long long)TMPF * TMPP);
    copy_strided<<<cdiv_ll((long long)TMPF * TMPF, B), B, 0, stream>>>(
        W2p, TMPP, W2, TMPF, TMPF, TMPF);

    // --- gcn_norm (once) ---
    zero_flat<<<cdiv_ll(NN, B), B, 0, stream>>>(deg, NN);
    deg_accum<<<cdiv_ll(EE, B), B, 0, stream>>>(ei, ew, deg, EE);
    dinv_kernel<<<cdiv_ll(NN, B), B, 0, stream>>>(deg, dinv, NN);
    norm_kernel<<<cdiv_ll(EE, B), B, 0, stream>>>(ei, ew, dinv, norm, EE);

    // --- x working copy (never mutate inputs) ---
    copy_flat<<<cdiv_ll((long long)NN * TPAST, B), B, 0, stream>>>(xb0, x, (long long)NN * TPAST);

    for (int t = 0; t < TFUT; ++t) {
        float* xcur  = (t & 1) ? xb1 : xb0;
        float* xnext = (t & 1) ? xb0 : xb1;
        float* h0 = temp + TPAST;          // cols 12..139 (ld = TMPF)
        float* h1 = temp + TPAST + HID;    // cols 140..267

        // temp[:,0:12] = x
        copy_strided<<<cdiv_ll((long long)NN * TPAST, B), B, 0, stream>>>(
            temp, TMPF, xcur, TPAST, NN, TPAST);

        // --- conv0: h0 = lrelu(scatter(x@W0)) ---
        {
            int tiles = (NN / 16) * (HID / 16);
            wmma_gemm_f32<TPAST, HID, TPAST, HID><<<cdiv_ll(tiles, B / 32), B, 0, stream>>>(
                xcur, W0, G, NN);
            zero_strided<<<cdiv_ll((long long)NN * HID, B), B, 0, stream>>>(h0, NN, HID, TMPF);
            edge_scatter_vec4<HID><<<cdiv_ll((long long)EE * (HID / 4), B), B, 0, stream>>>(
                G, HID, h0, TMPF, ei, norm, EE);
            finish_lrelu<HID><<<cdiv_ll((long long)NN * HID, B), B, 0, stream>>>(
                h0, TMPF, G, HID, dinv, b0, NN);
        }

        // --- conv1: h1 = lrelu(scatter(h0@W1)) ---
        {
            int tiles = (NN / 16) * (HID / 16);
            wmma_gemm_f32<HID, HID, TMPF, HID><<<cdiv_ll(tiles, B / 32), B, 0, stream>>>(
                h0, W1, G, NN);
            zero_strided<<<cdiv_ll((long long)NN * HID, B), B, 0, stream>>>(h1, NN, HID, TMPF);
            edge_scatter_vec4<HID><<<cdiv_ll((long long)EE * (HID / 4), B), B, 0, stream>>>(
                G, HID, h1, TMPF, ei, norm, EE);
            finish_lrelu<HID><<<cdiv_ll((long long)NN * HID, B), B, 0, stream>>>(
                h1, TMPF, G, HID, dinv, b1, NN);
        }

        // --- conv2: xt = lrelu(scatter(temp@W2p)) ; G has ld=272, cols 268..271 unused ---
        {
            int tiles = (NN / 16) * (TMPP / 16);
            wmma_gemm_f32<TMPF, TMPP, TMPF, TMPP><<<cdiv_ll(tiles, B / 32), B, 0, stream>>>(
                temp, W2p, G, NN);
            zero_strided<<<cdiv_ll((long long)NN * TMPF, B), B, 0, stream>>>(xt, NN, TMPF, TMPF);
            edge_scatter_vec4<TMPF><<<cdiv_ll((long long)EE * (TMPF / 4), B), B, 0, stream>>>(
                G, TMPP, xt, TMPF, ei, norm, EE);
            finish_lrelu<TMPF><<<cdiv_ll((long long)NN * TMPF, B), B, 0, stream>>>(
                xt, TMPF, G, TMPP, dinv, b2, NN);
        }

        // --- conv3: yp = scatter(xt@W3) + b3 (no lrelu) ---
        gemv268<<<cdiv_ll(NN, B), B, 0, stream>>>(xt, W3, g3, NN);
        zero_flat<<<cdiv_ll(NN, B), B, 0, stream>>>(deg, NN);          // deg == ypacc
        edge_scatter_scalar<<<cdiv_ll(EE, B), B, 0, stream>>>(g3, deg, ei, norm, EE);
        finalize_step<<<cdiv_ll(NN, B), B, 0, stream>>>(
            deg, g3, dinv, b3, xcur, xnext, out, NN, t);
    }
}